// GCNSampling_12618613915856
// MI455X (gfx1250) — compile-verified
//
#include <hip/hip_runtime.h>
#include <hip/hip_bf16.h>

typedef __attribute__((ext_vector_type(2))) float v2f;
typedef __attribute__((ext_vector_type(8))) float v8f;

#define FEAT 64  // K dimension for both layers (IN_FEATS == N_HIDDEN == 64)

// ---------------------------------------------------------------------------
// In-degree: one thread per edge, hardware f32 atomic add.
// ---------------------------------------------------------------------------
__global__ __launch_bounds__(256) void degree_kernel(const int* __restrict__ dst,
                                                     float* __restrict__ deg, int E) {
    int e = blockIdx.x * blockDim.x + threadIdx.x;
    if (e < E) {
        unsafeAtomicAdd(&deg[dst[e]], 1.0f);
    }
}

// ---------------------------------------------------------------------------
// Scatter-add aggregation: 64 consecutive threads handle one edge's 64 feats.
// Gather feat[src] row (coalesced 256B), atomic-add into agg[dst] row.
// ---------------------------------------------------------------------------
__global__ __launch_bounds__(256) void scatter_add_kernel(const float* __restrict__ feat,
                                                          const int* __restrict__ src,
                                                          const int* __restrict__ dst,
                                                          float* __restrict__ agg,
                                                          long long total) {
    long long i = (long long)blockIdx.x * blockDim.x + threadIdx.x;
    if (i >= total) return;
    int e = (int)(i >> 6);
    int f = (int)(i & 63);
    int s = src[e];
    int d = dst[e];
    float v = feat[(long long)s * FEAT + f];
    unsafeAtomicAdd(&agg[(long long)d * FEAT + f], v);
}

// ---------------------------------------------------------------------------
// WMMA GEMM tile kernel: one wave32 per 16x16 output tile.
//   out[row, col] = act( inv_deg[row] * (S[row,:] @ W[:,col]) + bias[col] )
// S: [n x 64] row-major, W: [64 x ncols] row-major.
// K=64 swept as 16 x V_WMMA_F32_16X16X4_F32.
// ---------------------------------------------------------------------------
__global__ __launch_bounds__(32) void wmma_gemm_kernel(const float* __restrict__ S,
                                                       const float* __restrict__ deg,
                                                       const float* __restrict__ W,
                                                       const float* __restrict__ bias,
                                                       float* __restrict__ out,
                                                       int n, int tilesN, int ncols,
                                                       int do_relu) {
    const int tile = blockIdx.x;
    const int tm = tile / tilesN;
    const int tn = tile % tilesN;
    const int lane = threadIdx.x;        // 0..31
    const int half = lane >> 4;          // 0 or 1
    const int l15  = lane & 15;

    // A-matrix 16x4 f32 layout: lane<16 -> K offsets {0,1}; lane>=16 -> {2,3}; M = lane&15
    int rowA = tm * 16 + l15;
    if (rowA >= n) rowA = n - 1;         // branch-free clamp (N%16==0 normally)
    const int kb = half * 2;

    // B-matrix 4x16 f32 layout (symmetric with A): N = lane&15, same K pairing
    const int colB = tn * 16 + l15;
    const int cB   = (colB < ncols) ? colB : 0;
    const float bmask = (colB < ncols) ? 1.0f : 0.0f;

    const float* Sp = S + (long long)rowA * FEAT;

    v8f c = {};
    #pragma unroll
    for (int k0 = 0; k0 < FEAT; k0 += 4) {
        const int k = k0 + kb;
        v2f a = *(const v2f*)(Sp + k);                 // S[rowA][k], S[rowA][k+1]
        v2f b;
        b.x = W[(long long)k * ncols + cB] * bmask;    // W[k][colB]
        b.y = W[(long long)(k + 1) * ncols + cB] * bmask;
        c = __builtin_amdgcn_wmma_f32_16x16x4_f32(
                /*neg_a=*/false, a, /*neg_b=*/false, b,
                /*c_mod=*/(short)0, c, /*reuse_a=*/false, /*reuse_b=*/false);
    }

    // Epilogue. C/D layout: VGPR r, lanes 0-15 -> M=r, N=lane; lanes 16-31 -> M=r+8.
    const float bv = (colB < ncols) ? bias[colB] : 0.0f;
    #pragma unroll
    for (int r = 0; r < 8; ++r) {
        int row = tm * 16 + r + half * 8;
        int rc = (row < n) ? row : (n - 1);
        float d = deg[rc];
        float inv = (d > 0.0f) ? (1.0f / d) : 0.0f;
        float v = c[r] * inv + bv;
        if (do_relu) v = fmaxf(v, 0.0f);
        if (colB < ncols && row < n) {
            out[(long long)row * ncols + colB] = v;
        }
    }
}

// ---------------------------------------------------------------------------
// Launch
// Inputs: [0] features (N*64 f32), [1] W1 (64*64), [2] b1 (64),
//         [3] W2 (64*40), [4] b2 (40), [5] src (E i32), [6] dst (E i32)
// Output: N*40 f32
// Workspace layout: agg [N*64 f32] | deg [N f32] | h1 [N*64 f32]
// ---------------------------------------------------------------------------
extern "C" void kernel_launch(void* const* d_in, const int* in_sizes, int n_in,
                              void* d_out, int out_size, void* d_ws, size_t ws_size,
                              hipStream_t stream) {
    const float* features = (const float*)d_in[0];
    const float* W1 = (const float*)d_in[1];
    const float* b1 = (const float*)d_in[2];
    const float* W2 = (const float*)d_in[3];
    const float* b2 = (const float*)d_in[4];
    const int* src = (const int*)d_in[5];
    const int* dst = (const int*)d_in[6];
    float* out = (float*)d_out;

    const int N = in_sizes[0] / FEAT;   // 100000
    const int E = in_sizes[5];          // 1200000
    const int NC1 = FEAT;               // hidden = 64
    const int NC2 = in_sizes[4];        // classes = 40

    float* agg = (float*)d_ws;
    float* deg = agg + (long long)N * FEAT;
    float* h1  = deg + N;

    // Zero agg + deg (contiguous)
    hipMemsetAsync(agg, 0, (size_t)((long long)N * FEAT + N) * sizeof(float), stream);

    // Degree
    degree_kernel<<<(E + 255) / 256, 256, 0, stream>>>(dst, deg, E);

    // Aggregate features -> agg
    long long total = (long long)E * FEAT;
    int blocks = (int)((total + 255) / 256);
    scatter_add_kernel<<<blocks, 256, 0, stream>>>(features, src, dst, agg, total);

    // Layer 1: h1 = relu(inv_deg * (agg @ W1) + b1)
    const int tilesM = (N + 15) / 16;   // 6250
    const int tilesN1 = (NC1 + 15) / 16; // 4
    wmma_gemm_kernel<<<tilesM * tilesN1, 32, 0, stream>>>(agg, deg, W1, b1, h1,
                                                          N, tilesN1, NC1, /*relu=*/1);

    // Re-zero agg, aggregate h1 -> agg
    hipMemsetAsync(agg, 0, (size_t)((long long)N * FEAT) * sizeof(float), stream);
    scatter_add_kernel<<<blocks, 256, 0, stream>>>(h1, src, dst, agg, total);

    // Layer 2: out = inv_deg * (agg @ W2) + b2
    const int tilesN2 = (NC2 + 15) / 16; // 3
    wmma_gemm_kernel<<<tilesM * tilesN2, 32, 0, stream>>>(agg, deg, W2, b2, out,
                                                          N, tilesN2, NC2, /*relu=*/0);
}